// DeepNCMDecoder_54133767799349
// MI455X (gfx1250) — compile-verified
//
#include <hip/hip_runtime.h>

typedef __attribute__((ext_vector_type(2))) float v2f;
typedef __attribute__((ext_vector_type(8))) float v8f;

#define NB 8192     // batch rows
#define NP 10000    // prototypes
#define ND 128      // encoding dim

#define BLK_M 128   // 8 waves x 16 rows
#define BLK_N 80    // 5 wmma tiles x 16 cols ; 10000 = 125 * 80
#define LDSS  132   // padded LDS row stride (floats): 132 % 64 = 4 -> conflict-free frag reads

// ---------------------------------------------------------------------------
// Kernel 1: row squared-norms for embedded (x2) and prototypes (p2) -> d_ws
// ---------------------------------------------------------------------------
__global__ __launch_bounds__(256) void row_norms_kernel(
    const float* __restrict__ emb, const float* __restrict__ protos,
    float* __restrict__ ws)
{
    int r = blockIdx.x * 256 + threadIdx.x;
    if (r < NB) {
        const float4* p = reinterpret_cast<const float4*>(emb + (size_t)r * ND);
        float s = 0.f;
        #pragma unroll 4
        for (int i = 0; i < ND / 4; ++i) {
            float4 v = p[i];
            s += v.x * v.x + v.y * v.y + v.z * v.z + v.w * v.w;
        }
        ws[r] = s;
    } else if (r < NB + NP) {
        int rp = r - NB;
        const float4* p = reinterpret_cast<const float4*>(protos + (size_t)rp * ND);
        float s = 0.f;
        #pragma unroll 4
        for (int i = 0; i < ND / 4; ++i) {
            float4 v = p[i];
            s += v.x * v.x + v.y * v.y + v.z * v.z + v.w * v.w;
        }
        ws[r] = s;
    }
}

// ---------------------------------------------------------------------------
// Kernel 2: zero the prototype_updates + counts region of d_out
// ---------------------------------------------------------------------------
__global__ __launch_bounds__(256) void zero_kernel(float* __restrict__ p, int n)
{
    int i = blockIdx.x * 256 + threadIdx.x;
    if (i < n) p[i] = 0.f;
}

// ---------------------------------------------------------------------------
// Kernel 3: segment-sum scatter (prototype_updates, counts) via f32 atomics
// ---------------------------------------------------------------------------
__global__ __launch_bounds__(256) void scatter_add_kernel(
    const float* __restrict__ emb, const int* __restrict__ labels,
    float* __restrict__ updates, float* __restrict__ counts)
{
    int idx = blockIdx.x * 256 + threadIdx.x;     // 0 .. NB*ND-1
    int i = idx >> 7;                             // row
    int d = idx & (ND - 1);                       // dim
    int l = labels[i];
    __hip_atomic_fetch_add(&updates[(size_t)l * ND + d], emb[idx],
                           __ATOMIC_RELAXED, __HIP_MEMORY_SCOPE_AGENT);
    if (d == 0)
        __hip_atomic_fetch_add(&counts[l], 1.0f,
                               __ATOMIC_RELAXED, __HIP_MEMORY_SCOPE_AGENT);
}

// ---------------------------------------------------------------------------
// Kernel 4: scores = -(max(x2 + p2 - 2*X.P^T, 0)) via V_WMMA_F32_16X16X4_F32
//   block tile 128M x 80N, 8 waves (16 rows each), 5 accum tiles per wave.
//   B tile (80 proto rows x 128 K) staged in LDS, shared by all waves.
// ---------------------------------------------------------------------------
__global__ __launch_bounds__(256) void ncm_scores_wmma_kernel(
    const float* __restrict__ emb, const float* __restrict__ protos,
    const float* __restrict__ x2, const float* __restrict__ p2,
    float* __restrict__ scores)
{
    __shared__ float ldsB[BLK_N * LDSS];

    const int tid    = threadIdx.x;
    const int lane   = tid & 31;
    const int wave   = tid >> 5;            // 0..7 -> M sub-block
    const int blockN = blockIdx.x;          // 0..124
    const int blockM = blockIdx.y;          // 0..63

    // ---- stage B tile: protos[blockN*80 .. +80) rows, full K=128 ----
    // 80*128 floats = 2560 float4, 256 threads -> 10 each (coalesced)
    {
        const float4* gsrc =
            reinterpret_cast<const float4*>(protos + (size_t)blockN * BLK_N * ND);
        #pragma unroll
        for (int i = 0; i < 10; ++i) {
            int lin = tid + i * 256;        // float4 index in tile
            int row = lin >> 5;             // / (128/4)
            int c4  = lin & 31;
            float4 v = gsrc[(size_t)row * (ND / 4) + c4];
            float* dst = &ldsB[row * LDSS + c4 * 4];
            dst[0] = v.x; dst[1] = v.y; dst[2] = v.z; dst[3] = v.w;
        }
    }
    __syncthreads();

    const int mlane = lane & 15;            // M (for A) / N (for B) within tile
    const int khalf = (lane >> 4) * 2;      // K sub-offset per half-wave

    // A fragment source: wave-private 16 rows of embedded, straight from L2
    const float* aRow =
        emb + (size_t)(blockM * BLK_M + wave * 16 + mlane) * ND;

    v8f acc[5];
    #pragma unroll
    for (int t = 0; t < 5; ++t)
        acc[t] = (v8f){0.f, 0.f, 0.f, 0.f, 0.f, 0.f, 0.f, 0.f};

    #pragma unroll 4
    for (int k = 0; k < ND; k += 4) {
        v2f aF = *reinterpret_cast<const v2f*>(aRow + k + khalf);
        #pragma unroll
        for (int t = 0; t < 5; ++t) {
            const float* bp = &ldsB[(t * 16 + mlane) * LDSS + k + khalf];
            v2f bF = *reinterpret_cast<const v2f*>(bp);
            acc[t] = __builtin_amdgcn_wmma_f32_16x16x4_f32(
                false, aF, false, bF, (short)0, acc[t], false, false);
        }
    }

    // ---- epilogue: scores = -max(x2[m] + p2[n] - 2*acc, 0) ----
    // C/D layout: VGPR j -> M = j + (lane>=16 ? 8 : 0), N = lane & 15
    const int rowLo = blockM * BLK_M + wave * 16 + ((lane >> 4) * 8);
    float x2v[8];
    #pragma unroll
    for (int j = 0; j < 8; ++j) x2v[j] = x2[rowLo + j];

    #pragma unroll
    for (int t = 0; t < 5; ++t) {
        const int col = blockN * BLK_N + t * 16 + mlane;
        const float p2v = p2[col];
        #pragma unroll
        for (int j = 0; j < 8; ++j) {
            float d2 = fmaxf(x2v[j] + p2v - 2.0f * acc[t][j], 0.0f);
            scores[(size_t)(rowLo + j) * NP + col] = -d2;
        }
    }
}

// ---------------------------------------------------------------------------
extern "C" void kernel_launch(void* const* d_in, const int* in_sizes, int n_in,
                              void* d_out, int out_size, void* d_ws, size_t ws_size,
                              hipStream_t stream)
{
    const float* emb    = (const float*)d_in[0];   // [8192,128] f32
    const int*   labels = (const int*)d_in[1];     // [8192] i32
    const float* protos = (const float*)d_in[2];   // [10000,128] f32

    float* out     = (float*)d_out;
    float* scores  = out;                                  // [8192,10000]
    float* updates = out + (size_t)NB * NP;                // [10000,128]
    float* counts  = updates + (size_t)NP * ND;            // [10000]

    float* x2 = (float*)d_ws;                              // [8192]
    float* p2 = x2 + NB;                                   // [10000]

    // norms for the distance epilogue
    row_norms_kernel<<<(NB + NP + 255) / 256, 256, 0, stream>>>(emb, protos, x2);

    // zero + scatter segment sums (deterministic each call)
    int acc_n = NP * ND + NP;
    zero_kernel<<<(acc_n + 255) / 256, 256, 0, stream>>>(updates, acc_n);
    scatter_add_kernel<<<(NB * ND) / 256, 256, 0, stream>>>(emb, labels, updates, counts);

    // main WMMA distance kernel: grid 125 x 64 covers 10000 x 8192 exactly
    dim3 grid(NP / BLK_N, NB / BLK_M);
    ncm_scores_wmma_kernel<<<grid, 256, 0, stream>>>(emb, protos, x2, p2, scores);
}